// WES_55894704390236
// MI455X (gfx1250) — compile-verified
//
#include <hip/hip_runtime.h>
#include <cmath>

// ---- types ----
typedef __attribute__((ext_vector_type(2))) float v2f;
typedef __attribute__((ext_vector_type(4))) float f32x4;
typedef __attribute__((ext_vector_type(8))) float v8f;

#define BLOCK 256
#define WAVES (BLOCK / 32)
#define NBLK_MAX 1024

// ---- per-element weighted squared error (0.5 and 1/N folded into final scale) ----
__device__ __forceinline__ float weighted_se(float o, float t, float lgs) {
  // lgs = lgamma(EST_A) + log(EST_SCALE), computed on host
  const float LOC        = -1.1328205299926424e-27f;
  const float INV_SCALE  = (float)(1.0 / 1.5376362609160314);
  const float AM1        = (float)(0.06444291424643259 - 1.0);  // a - 1
  const float INV_FXMAX  = (float)(1.0 / 1.0500746950269468e+24);
  const float C          = (float)(1.0 / 107.2185);             // 1/(Y_MAX - Y_MIN)
  const float BMC        = 5.0f - C;                            // BETA - c

  float x   = (t - LOC) * INV_SCALE;
  float lp  = fmaf(AM1, __logf(x), -x) - lgs;   // (a-1)ln x - x - lgamma(a) - ln(scale)
  float pdf = __expf(lp);
  float ef  = fmaf(BMC, 1.0f - pdf * INV_FXMAX, C);
  float d   = o - t;
  return d * d * ef;
}

// Full wave32 sum with two V_WMMA_F32_16X16X4_F32, no cross-lane shuffles:
//   WMMA1: A = partials (A[m][0]=p_m, A[m][2]=p_{m+16}), B = ones
//          -> D[m][n] = R[m] = p_m + p_{m+16}
//          lanes 0-15 see R[0..7] in d[0..7]; lanes 16-31 see R[8..15].
//          s = sum(d[0..7])  => S_lo on low half, S_hi on high half.
//   WMMA2: A = ones, B[0][n]=S_lo, B[2][n]=S_hi
//          -> D[m][n] = S_lo + S_hi = wave total, identical in every lane.
__device__ __forceinline__ float wave_sum_wmma(float p) {
  v2f ones; ones.x = 1.0f; ones.y = 1.0f;
  v8f c0 = {};

  v2f a1; a1.x = p; a1.y = 0.0f;         // partials into A rows (K=0 / K=2)
  v8f d1 = __builtin_amdgcn_wmma_f32_16x16x4_f32(
      false, a1, false, ones, (short)0, c0, false, false);

  float s = d1[0] + d1[1] + d1[2] + d1[3] + d1[4] + d1[5] + d1[6] + d1[7];

  v2f b2; b2.x = s; b2.y = 0.0f;         // half-sums into B rows (K=0 / K=2)
  v8f d2 = __builtin_amdgcn_wmma_f32_16x16x4_f32(
      false, ones, false, b2, (short)0, c0, false, false);

  return d2[0];                           // wave total, every lane
}

__global__ __launch_bounds__(BLOCK)
void WES_gammaloss_partial(const float* __restrict__ out,
                           const float* __restrict__ tgt,
                           float* __restrict__ partials,
                           int n, float lgs, int nblk) {
  __shared__ float smem[WAVES];
  const int tid    = threadIdx.x;
  const int gid    = blockIdx.x * BLOCK + tid;
  const int stride = nblk * BLOCK;
  const int n4     = n >> 2;

  const f32x4* o4 = (const f32x4*)out;
  const f32x4* t4 = (const f32x4*)tgt;

  float acc = 0.0f;
  int i = gid;
  // 2x unrolled grid-stride: 4 NT B128 loads in flight per iteration
  for (; i + stride < n4; i += 2 * stride) {
    f32x4 ov0 = __builtin_nontemporal_load(o4 + i);
    f32x4 tv0 = __builtin_nontemporal_load(t4 + i);
    f32x4 ov1 = __builtin_nontemporal_load(o4 + i + stride);
    f32x4 tv1 = __builtin_nontemporal_load(t4 + i + stride);
    acc += weighted_se(ov0.x, tv0.x, lgs);
    acc += weighted_se(ov0.y, tv0.y, lgs);
    acc += weighted_se(ov0.z, tv0.z, lgs);
    acc += weighted_se(ov0.w, tv0.w, lgs);
    acc += weighted_se(ov1.x, tv1.x, lgs);
    acc += weighted_se(ov1.y, tv1.y, lgs);
    acc += weighted_se(ov1.z, tv1.z, lgs);
    acc += weighted_se(ov1.w, tv1.w, lgs);
  }
  for (; i < n4; i += stride) {
    f32x4 ov = __builtin_nontemporal_load(o4 + i);
    f32x4 tv = __builtin_nontemporal_load(t4 + i);
    acc += weighted_se(ov.x, tv.x, lgs);
    acc += weighted_se(ov.y, tv.y, lgs);
    acc += weighted_se(ov.z, tv.z, lgs);
    acc += weighted_se(ov.w, tv.w, lgs);
  }
  // scalar tail (n % 4), block 0 only
  const int tail = n & 3;
  if (blockIdx.x == 0 && tid < tail) {
    int idx = (n4 << 2) + tid;
    acc += weighted_se(out[idx], tgt[idx], lgs);
  }

  // control flow reconverged here: EXEC all-ones for the WMMAs
  float ws = wave_sum_wmma(acc);
  if ((tid & 31) == 0) smem[tid >> 5] = ws;
  __syncthreads();
  if (tid == 0) {
    float s = 0.0f;
#pragma unroll
    for (int w = 0; w < WAVES; ++w) s += smem[w];
    partials[blockIdx.x] = s;
  }
}

__global__ __launch_bounds__(BLOCK)
void WES_gammaloss_final(const float* __restrict__ partials,
                         float* __restrict__ result,
                         int nblk, float final_scale) {
  __shared__ float smem[WAVES];
  const int tid = threadIdx.x;
  float acc = 0.0f;
  for (int i = tid; i < nblk; i += BLOCK) acc += partials[i];
  float ws = wave_sum_wmma(acc);
  if ((tid & 31) == 0) smem[tid >> 5] = ws;
  __syncthreads();
  if (tid == 0) {
    float s = 0.0f;
#pragma unroll
    for (int w = 0; w < WAVES; ++w) s += smem[w];
    result[0] = s * final_scale;   // * 0.5 / N
  }
}

extern "C" void kernel_launch(void* const* d_in, const int* in_sizes, int n_in,
                              void* d_out, int out_size, void* d_ws, size_t ws_size,
                              hipStream_t stream) {
  const float* out = (const float*)d_in[0];
  const float* tgt = (const float*)d_in[1];
  float* result    = (float*)d_out;
  float* partials  = (float*)d_ws;
  const int n = in_sizes[0];

  int nblk = NBLK_MAX;
  if (ws_size < (size_t)nblk * sizeof(float)) {
    nblk = (int)(ws_size / sizeof(float));
    if (nblk < 1) nblk = 1;
  }

  // lgamma(EST_A) + log(EST_SCALE), folded on host for exactness
  const float lgs = (float)(lgamma(0.06444291424643259) + log(1.5376362609160314));
  const float final_scale = 0.5f / (float)n;

  WES_gammaloss_partial<<<nblk, BLOCK, 0, stream>>>(out, tgt, partials, n, lgs, nblk);
  WES_gammaloss_final<<<1, BLOCK, 0, stream>>>(partials, result, nblk, final_scale);
}